// PolyMinMatchingLoss_40200893890957
// MI455X (gfx1250) — compile-verified
//
#include <hip/hip_runtime.h>
#include <float.h>
#include <math.h>

// Chamfer/min-matching loss, B=8, P=4096, D=2.
// Inputs total 512KB -> fully WGP$/L2-resident on MI455X; the work is the
// 8*4096*4096 pairwise d^2 evaluations + axis-min reductions.
// Cross term <pred_n, gt_m> runs on the matrix pipe (V_WMMA_F32_16X16X4_F32,
// K padded 2->4 with zeros); d^2 epilogue uses packed f32 VALU; column mins
// accumulate in LDS via ds_min_u32 (branch-free, EXEC stays all-ones for WMMA);
// clamp per d2; sqrt deferred past the min (monotonic), so only 64K sqrts at the end.

typedef __attribute__((ext_vector_type(2))) float v2f;
typedef __attribute__((ext_vector_type(8))) float v8f;

#define NBATCH 8
#define NP     4096
#define EPS_F  1e-12f
#define ROWS_PER_BLOCK 128   // 8 waves * 16 pred rows
#define NTHREADS 256

// ---------------------------------------------------------------- init
__global__ __launch_bounds__(256) void cd_init(unsigned* __restrict__ colmin) {
    int i = blockIdx.x * blockDim.x + threadIdx.x;
    if (i < NBATCH * NP) colmin[i] = __float_as_uint(FLT_MAX);
}

// ---------------------------------------------------------------- main tiles
// grid: NBATCH * (NP / ROWS_PER_BLOCK) = 256 blocks, 256 threads (8 waves).
// Each wave: one 16-row pred tile, sweeps all NP/16 gt tiles with WMMA.
__global__ __launch_bounds__(NTHREADS) void cd_tiles(const float* __restrict__ pred,
                                                     const float* __restrict__ gt,
                                                     unsigned* __restrict__ colmin, // min over pred n, per (b,m)
                                                     float* __restrict__ rowmin)    // min over gt m,  per (b,n)
{
    __shared__ unsigned ldsCol[NP];   // 16KB: block-local col-min (per gt point m)

    const int b         = blockIdx.x >> 5;          // 32 pred groups per batch
    const int predGroup = blockIdx.x & 31;
    const int wave      = threadIdx.x >> 5;
    const int lane      = threadIdx.x & 31;
    const int hi        = lane >> 4;                // half-wave: 0 or 1
    const int l15       = lane & 15;

    const int rowBase = predGroup * ROWS_PER_BLOCK + wave * 16;  // pred index base n
    const float2* predB = (const float2*)(pred + (size_t)b * NP * 2);
    const float2* gtB   = (const float2*)(gt   + (size_t)b * NP * 2);

    // Block-local col-min init.
    for (int i = threadIdx.x; i < NP; i += NTHREADS)
        ldsCol[i] = __float_as_uint(FLT_MAX);

    // One-shot prefetch of this batch's gt array into the WGP$:
    // 4096 * 8B = 32KB = 256 cachelines, one per thread.  -> global_prefetch_b8
    __builtin_prefetch((const void*)(gtB + threadIdx.x * 16), 0, 3);

    // A operand (16x4 f32, 2 VGPRs): lane holds pred row M = lane&15;
    // VGPR0/1 = K0/K1 in lanes 0-15, K2/K3 (zero pad) in lanes 16-31.
    v2f a;
    {
        float2 p = predB[rowBase + l15];
        a.x = hi ? 0.0f : p.x;
        a.y = hi ? 0.0f : p.y;
    }

    // ||pred||^2 per C-element position: C elem (v,lane) is row M = v + 8*hi.
    float p2v[8];
#pragma unroll
    for (int v = 0; v < 8; ++v) {
        float2 p = predB[rowBase + v + 8 * hi];
        p2v[v] = p.x * p.x + p.y * p.y;
    }

    float rmin[8];
#pragma unroll
    for (int v = 0; v < 8; ++v) rmin[v] = FLT_MAX;

    __syncthreads();   // ldsCol ready

    // Software-pipelined sweep over all gt tiles (load tile mt+1 while
    // computing tile mt, so s_wait_loadcnt overlaps the WMMA + epilogue).
    float2 g = gtB[l15];
    for (int mt = 0; mt < NP / 16; ++mt) {
        const float2 gn = gtB[(((mt + 1) & (NP / 16 - 1)) << 4) + l15];

        const float g2 = g.x * g.x + g.y * g.y;

        // B operand (4x16 f32, 2 VGPRs): rows striped across lanes;
        // rows 0/1 = gt x/y in lanes 0-15, rows 2/3 = zero pad in lanes 16-31.
        v2f bb;
        bb.x = hi ? 0.0f : g.x;
        bb.y = hi ? 0.0f : g.y;

        v8f c = {};
        c = __builtin_amdgcn_wmma_f32_16x16x4_f32(
                /*neg_a=*/false, a, /*neg_b=*/false, bb,
                /*c_mod=*/(short)0, c, /*reuse_a=*/false, /*reuse_b=*/false);

        float cmin = FLT_MAX;
#pragma unroll
        for (int v = 0; v < 8; ++v) {
            float d2 = fmaxf(p2v[v] + g2 - 2.0f * c[v], EPS_F);  // clamp -> all > 0
            rmin[v] = fminf(rmin[v], d2);   // running min over m (this lane's N column)
            cmin    = fminf(cmin, d2);      // min over this tile's 8 pred rows (this half)
        }
        // Lanes L and L+16 see the same gt point m: both ds_min the same slot
        // (merges half-waves for free; no shuffle, no divergence, EXEC all-ones).
        atomicMin(&ldsCol[(mt << 4) + l15], __float_as_uint(cmin));

        g = gn;
    }

    // Reduce rmin across the 16 lanes of each half (min over all N columns),
    // then one lane per half stores the completed gt2pred min (full m coverage).
#pragma unroll
    for (int v = 0; v < 8; ++v) {
        float r = rmin[v];
        r = fminf(r, __shfl_xor(r, 1, 32));
        r = fminf(r, __shfl_xor(r, 2, 32));
        r = fminf(r, __shfl_xor(r, 4, 32));
        r = fminf(r, __shfl_xor(r, 8, 32));
        if (l15 == 0)
            rowmin[(size_t)b * NP + rowBase + v + 8 * hi] = r;
    }

    // Block-level merge of col-min into global (8x fewer global atomics than
    // per-wave updates; all d2 >= EPS > 0 so uint order == float order).
    __syncthreads();
    for (int i = threadIdx.x; i < NP; i += NTHREADS)
        atomicMin(&colmin[(size_t)b * NP + i], ldsCol[i]);
}

// ---------------------------------------------------------------- reduce
__global__ __launch_bounds__(1024) void cd_reduce(const unsigned* __restrict__ colmin,
                                                  const float* __restrict__ rowmin,
                                                  float* __restrict__ out)
{
    __shared__ double s0[1024];
    __shared__ double s1[1024];
    double a0 = 0.0, a1 = 0.0;
    for (int i = threadIdx.x; i < NBATCH * NP; i += 1024) {
        a0 += (double)sqrtf(__uint_as_float(colmin[i]));  // pred2gt_min
        a1 += (double)sqrtf(rowmin[i]);                   // gt2pred_min
    }
    s0[threadIdx.x] = a0;
    s1[threadIdx.x] = a1;
    __syncthreads();
    for (int off = 512; off > 0; off >>= 1) {
        if (threadIdx.x < off) {
            s0[threadIdx.x] += s0[threadIdx.x + off];
            s1[threadIdx.x] += s1[threadIdx.x + off];
        }
        __syncthreads();
    }
    if (threadIdx.x == 0) {
        const double n = (double)(NBATCH * NP);
        out[0] = (float)(0.5 * (s0[0] / n + s1[0] / n));
    }
}

// ---------------------------------------------------------------- launch
extern "C" void kernel_launch(void* const* d_in, const int* in_sizes, int n_in,
                              void* d_out, int out_size, void* d_ws, size_t ws_size,
                              hipStream_t stream) {
    (void)in_sizes; (void)n_in; (void)out_size; (void)ws_size;
    const float* pred = (const float*)d_in[0];
    const float* gt   = (const float*)d_in[1];
    float* out = (float*)d_out;

    unsigned* colmin = (unsigned*)d_ws;                               // B*P uints
    float*    rowmin = (float*)((char*)d_ws + sizeof(unsigned) * NBATCH * NP);

    cd_init<<<(NBATCH * NP + 255) / 256, 256, 0, stream>>>(colmin);
    cd_tiles<<<NBATCH * (NP / ROWS_PER_BLOCK), NTHREADS, 0, stream>>>(pred, gt, colmin, rowmin);
    cd_reduce<<<1, 1024, 0, stream>>>(colmin, rowmin, out);
}